// SpaceToPatch_position_88210038325811
// MI455X (gfx1250) — compile-verified
//
#include <hip/hip_runtime.h>

// ---------------------------------------------------------------------------
// SpaceToPatch position-correct + space-to-depth masked-normalize reduction.
// Shapes fixed by the reference: x[16,64,256,256] f32 -> out[16,1,64,64] f32.
// Bandwidth-bound: 256 MB single-pass read @ 23.3 TB/s  => ~11 us floor.
//  - NT b128 loads (dataset > 192 MB L2, zero reuse)
//  - global_prefetch_b8 ahead on the 256KB-strided channel walk
//  - v_wmma_f32_16x16x4_f32 (ones-B) for the 4-way partial reduction tree
// ---------------------------------------------------------------------------

typedef float v2f __attribute__((ext_vector_type(2)));
typedef float v4f __attribute__((ext_vector_type(4)));
typedef float v8f __attribute__((ext_vector_type(8)));

#define NB 16
#define NC 64
#define NH 256
#define NW 256
#define BS 4
#define DH (NH / BS) // 64
#define DW (NW / BS) // 64

__global__ __launch_bounds__(256)
void s2p_masked_norm_kernel(const float* __restrict__ in,
                            const int* __restrict__ dirp,
                            float* __restrict__ out) {
    const int t   = threadIdx.x;
    const int dw  = t & 63;     // output column handled by this lane
    const int g   = t >> 6;     // 0..3 == bh (row inside the 4x4 block)
    const int blk = blockIdx.x; // 0..1023
    const int b   = blk >> 6;
    const int dhi = blk & 63;
    const int dir = *dirp;      // 1: add h%4 where x<0 ; 2: add w%4

    __shared__ float pN[BS][DW];   // per-group partial numerators
    __shared__ float pD[BS][DW];   // per-group partial denominators
    __shared__ float red[2 * DW];  // WMMA-reduced sums: [0,64)=num [64,128)=den

    const int    h    = dhi * BS + g;
    const size_t cs   = (size_t)NH * NW;                // channel stride (floats)
    const float* base = in + (size_t)b * NC * cs + (size_t)h * NW + (size_t)dw * 4;

    float num = 0.0f, den = 0.0f;
    const float corrH = (float)g;   // h % 4 == bh, constant per thread

    #pragma unroll 8
    for (int c = 0; c < NC; ++c) {
        const float* p = base + (size_t)c * cs;
        // speculative prefetch 2 channels (512 KB) ahead; OOB tail is dropped
        __builtin_prefetch(p + 2 * cs, 0, 0);
        // single-use stream: non-temporal 128-bit load
        v4f v = __builtin_nontemporal_load(reinterpret_cast<const v4f*>(p));
        #pragma unroll
        for (int j = 0; j < 4; ++j) {
            const float corr = (dir == 1) ? corrH : (float)j; // w%4 == j
            float y = v[j];
            if (y < 0.0f) y += corr;            // correct_position
            num += y;                           // sum
            den += (y > 0.0f) ? 1.0f : y;       // masked count/sum
        }
    }

    pN[g][dw] = num;
    pD[g][dw] = den;
    __syncthreads();

    // ---- 4-way partial reduction on the matrix pipe: D = A x ones ----------
    // 128 quantities (q<64 -> num[dw], q>=64 -> den[dw-64]), 4 partials each.
    // A (16x4 f32): lanes 0-15 hold K=0(a0)/K=1(a1); lanes 16-31 K=2/K=3.
    // With B == ones and C == 0: D[m][n] = sum_k A[m][k] for every n.
    if (t < 32) {
        const int m  = t & 15;
        const int kb = (t < 16) ? 0 : 2;
        v2f ones; ones[0] = 1.0f; ones[1] = 1.0f;
        #pragma unroll
        for (int tile = 0; tile < 8; ++tile) {
            const int q0 = tile * 16;            // q uniform-side per tile
            const int q  = q0 + m;
            const float* src = (q0 < 64) ? &pN[0][0] : &pD[0][0];
            const int qq = (q0 < 64) ? q : (q - 64);
            v2f a;
            a[0] = src[(kb)     * DW + qq];
            a[1] = src[(kb + 1) * DW + qq];
            v8f cacc = {};
            v8f d = __builtin_amdgcn_wmma_f32_16x16x4_f32(
                false, a, false, ones, (short)0, cacc, false, false);
            // D layout: VGPR r, lane 0  -> D[r][0]   = S[q0 + r]
            //                   lane 16 -> D[8+r][0] = S[q0 + 8 + r]
            if (m == 0) {
                const int roff = (t < 16) ? 0 : 8;
                #pragma unroll
                for (int r = 0; r < 8; ++r)
                    red[q0 + roff + r] = d[r];
            }
        }
    }
    __syncthreads();

    if (t < DW) {
        const float o = red[t] / (red[DW + t] + 1.0f);
        out[(size_t)b * DH * DW + (size_t)dhi * DW + t] = o;
    }
}

extern "C" void kernel_launch(void* const* d_in, const int* in_sizes, int n_in,
                              void* d_out, int out_size, void* d_ws, size_t ws_size,
                              hipStream_t stream) {
    (void)in_sizes; (void)n_in; (void)d_ws; (void)ws_size; (void)out_size;
    const float* x   = (const float*)d_in[0];
    const int*   dir = (const int*)d_in[1];
    float*       out = (float*)d_out;
    // 16 batches x 64 dh rows = 1024 workgroups of 256 threads (8 wave32s)
    s2p_masked_norm_kernel<<<NB * DH, 256, 0, stream>>>(x, dir, out);
}